// BayesianTorchModel_37022618092110
// MI455X (gfx1250) — compile-verified
//
#include <hip/hip_runtime.h>
#include <stdint.h>

typedef __attribute__((ext_vector_type(4))) unsigned int v4u;
typedef __attribute__((ext_vector_type(8))) int          v8i;
typedef __attribute__((ext_vector_type(4))) int          v4i;

#define MBLK 256   // CPT rows per block (16 KB LDS tile)
#define BBLK 8     // batch values per block (all share one 32-bit pack word)

// ---- Pass 1: bit-pack evidence: pack[w][node] bit j = evidence[w*32+j][node] ----
__global__ void pack_evidence(const int* __restrict__ ev, uint32_t* __restrict__ pack,
                              int B, int N) {
  int node = blockIdx.x * blockDim.x + threadIdx.x;   // lanes -> consecutive nodes (coalesced)
  int w    = blockIdx.y;
  if (node >= N) return;
  uint32_t word = 0;
  int base_b = w * 32;
#pragma unroll 4
  for (int j = 0; j < 32; ++j) {
    int b = base_b + j;
    if (b < B) {
      uint32_t bit = (uint32_t)ev[(size_t)b * N + node] & 1u;
      word |= bit << j;
    }
  }
  pack[(size_t)w * N + node] = word;                  // 1.28 MB total, L2-resident afterwards
}

// ---- Roots: out[b, r] = sigmoid(root_logits[r]) ----
__global__ void roots_kernel(const float* __restrict__ root_logits, float* __restrict__ out,
                             int B, int N, int R) {
  int t = blockIdx.x * blockDim.x + threadIdx.x;
  if (t >= B * R) return;
  int b = t / R, r = t - b * R;
  float s = 1.0f / (1.0f + __expf(-root_logits[r]));
  __builtin_nontemporal_store(s, &out[(size_t)b * N + r]);
}

// ---- Main pass: TDM-stage CPT chunk to LDS, gather conf bits, sigmoid, NT store ----
__global__ void cpt_kernel(const int* __restrict__ parents,
                           const float* __restrict__ cpt,
                           const uint32_t* __restrict__ pack,
                           float* __restrict__ out,
                           int B, int N, int M, int R) {
  __shared__ float lds_cpt[MBLK * 16];                // 16 KB
  const int chunk = blockIdx.x;
  const int m0    = chunk * MBLK;
  const int rows  = min(MBLK, M - m0);
  const int b0    = blockIdx.y * BBLK;

  // Wave 0 issues one Tensor-DMA of this chunk's CPT rows (rows x 16 f32, contiguous).
  if (threadIdx.x < 32) {
    uint64_t ga     = (uint64_t)(uintptr_t)(cpt + (size_t)m0 * 16);
    uint32_t ldsoff = (uint32_t)(uintptr_t)&lds_cpt[0];   // low 32 bits of LDS flat addr = LDS offset
    uint32_t elems  = (uint32_t)rows * 16u;               // <= 4096, fits tile_dim0[15:0]
    v4u g0;
    g0.x = 1u;                                            // count=1 (valid descriptor)
    g0.y = ldsoff;                                        // lds_addr
    g0.z = (uint32_t)ga;                                  // global_addr[31:0]
    g0.w = ((uint32_t)(ga >> 32) & 0x01FFFFFFu) | (2u << 30); // global_addr[56:32] | type=2
    v8i g1;
    g1[0] = (int)(2u << 16);                              // workgroup_mask=0, data_size=4B
    g1[1] = (int)((elems & 0xFFFFu) << 16);               // tensor_dim0[15:0] @ bits[63:48]
    g1[2] = (int)(((elems >> 16) & 0xFFFFu) | (1u << 16));// tensor_dim0[31:16] | tensor_dim1=1
    g1[3] = (int)((elems & 0xFFFFu) << 16);               // tensor_dim1 hi=0 | tile_dim0=elems
    g1[4] = 0;                                            // tile_dim1=0, tile_dim2=0 (unused)
    g1[5] = (int)elems;                                   // tensor_dim0_stride (benign, single row)
    g1[6] = 0;
    g1[7] = 0;
    v4i g2; g2[0] = 1; g2[1] = 1; g2[2] = 0; g2[3] = 0;   // tensor_dim2=1, tensor_dim3=1
    v4i g3; g3[0] = 0; g3[1] = (int)(1u << 16); g3[2] = 0; g3[3] = 0; // tensor_dim4=1
    v8i g4; g4[0] = 0; g4[1] = 0; g4[2] = 0; g4[3] = 0;   // extra operand (zero-filled, per
    g4[4] = 0; g4[5] = 0; g4[6] = 0; g4[7] = 0;           //  6-arg clang-23 signature)
    __builtin_amdgcn_tensor_load_to_lds(g0, g1, g2, g3, g4, 0);
    __builtin_amdgcn_s_wait_tensorcnt(0);
  }
  __syncthreads();

  const int m_local = threadIdx.x;
  const int m = m0 + m_local;
  if (m >= M) return;

  int4 p = ((const int4*)parents)[m];                 // 16B per thread, coalesced
  const int w = b0 >> 5;                              // same pack word group for all BBLK batches
  const uint32_t* __restrict__ pw = pack + (size_t)w * N;
  uint32_t b3 = pw[p.x], b2 = pw[p.y], b1 = pw[p.z], bLSB = pw[p.w]; // 4 L2-hot gathers / 8 outputs

  const int nb = min(BBLK, B - b0);
  for (int bi = 0; bi < nb; ++bi) {
    int b   = b0 + bi;
    int bit = b & 31;
    uint32_t conf = (((b3   >> bit) & 1u) << 3)
                  | (((b2   >> bit) & 1u) << 2)
                  | (((b1   >> bit) & 1u) << 1)
                  |  ((bLSB >> bit) & 1u);
    float x = lds_cpt[m_local * 16 + (int)conf];
    float s = 1.0f / (1.0f + __expf(-x));
    // lanes -> consecutive m: coalesced; streamed once: non-temporal
    __builtin_nontemporal_store(s, &out[(size_t)b * N + R + m]);
  }
}

extern "C" void kernel_launch(void* const* d_in, const int* in_sizes, int n_in,
                              void* d_out, int out_size, void* d_ws, size_t ws_size,
                              hipStream_t stream) {
  const int*   ev          = (const int*)d_in[0];
  const int*   parents     = (const int*)d_in[1];
  const float* root_logits = (const float*)d_in[2];
  const float* cpt         = (const float*)d_in[3];
  float*       out         = (float*)d_out;

  const int R = in_sizes[2];          // 4 roots
  const int M = in_sizes[1] / 4;      // 9996
  const int N = M + R;                // 10000
  const int B = in_sizes[0] / N;      // 1024
  const int W = (B + 31) / 32;        // 32 pack words per node

  uint32_t* pack = (uint32_t*)d_ws;   // needs W*N*4 = 1.28 MB

  dim3 gp((N + 255) / 256, W);
  pack_evidence<<<gp, 256, 0, stream>>>(ev, pack, B, N);

  roots_kernel<<<(B * R + 255) / 256, 256, 0, stream>>>(root_logits, out, B, N, R);

  dim3 gc((M + MBLK - 1) / MBLK, (B + BBLK - 1) / BBLK);
  cpt_kernel<<<gc, MBLK, 0, stream>>>(parents, cpt, pack, out, B, N, M, R);
}